// FasterRCNNDetector_39152921870476
// MI455X (gfx1250) — compile-verified
//
#include <hip/hip_runtime.h>
#include <cmath>

// ---------------------------------------------------------------------------
// Faster R-CNN detection head for MI455X (gfx1250, wave32).
// GEMM is HBM-bound (AI ~5 FLOP/B over 262MB of feats) -> full-precision
// V_WMMA_F32_16X16X4_F32 is free; N_out=10 padded to one 16-wide WMMA tile.
// ---------------------------------------------------------------------------

typedef __attribute__((ext_vector_type(2))) float v2f;
typedef __attribute__((ext_vector_type(8))) float v8f;

#define BATCH   16
#define NPROP   4000
#define DIM     1024
#define NOUT    10                       // 2 cls logits + 8 box deltas
#define IMG     800.0f
#define SCORE_T 0.25f
#define NMS_T   0.5f
#define MAXDET  100
#define CLIPV   4.135166556742356f       // log(1000/16)

// ---------------------------------------------------------------------------
// Kernel 1: fused GEMM (WMMA f32 16x16x4) + softmax + box decode + threshold.
// grid = 1000 blocks x 128 threads (4 wave32s); one wave per 16-row tile.
// ---------------------------------------------------------------------------
__global__ __launch_bounds__(128) void det_head_kernel(
    const float* __restrict__ feats,      // (B*N, 1024)
    const float* __restrict__ proposals,  // (B*N, 4)
    const float* __restrict__ Wc,         // (2, 1024)
    const float* __restrict__ bc,         // (2,)
    const float* __restrict__ Wb,         // (8, 1024)
    const float* __restrict__ bb,         // (8,)
    float*       __restrict__ out,        // (B*N, 5): [0:4]=boxes, [4] later
    float*       __restrict__ ws_scores)  // (B*N,) thresholded scores
{
    __shared__ float Wlds[NOUT * DIM];     // 40 KB combined weight (rows 0-1 cls, 2-9 box)
    __shared__ float tileLds[4][16 * 16];  // per-wave accumulator transpose scratch

    const int tid = threadIdx.x;

    // Cooperative float4 staging of the combined 10x1024 weight matrix.
    for (int i4 = tid; i4 < (NOUT * DIM) / 4; i4 += blockDim.x) {
        const int row = i4 >> 8;          // 256 float4 per 1024-wide row
        const int c4  = i4 & 255;
        const float4* src = (row < 2)
            ? ((const float4*)(Wc + (size_t)row * DIM) + c4)
            : ((const float4*)(Wb + (size_t)(row - 2) * DIM) + c4);
        ((float4*)Wlds)[i4] = *src;
    }
    __syncthreads();

    const int wave = tid >> 5;
    const int lane = tid & 31;
    const int half = lane >> 4;           // 0 -> K pair {0,1}, 1 -> K pair {2,3}
    const int col  = lane & 15;           // output-channel column / row-in-tile

    const int  tile = blockIdx.x * 4 + wave;   // 4000 tiles of 16 rows total
    const long row0 = (long)tile * 16;

    // A fragment: lane (half,col) holds feats[row0+col][k + 2*half .. +1]
    const float* arow = feats + (row0 + col) * (long)DIM;
    // B fragment: B[k][n] = W[n][k]; cols >= NOUT are dont-care (clamped to row 0;
    // those accumulator columns are never read).
    const float* brow = Wlds + (size_t)(col < NOUT ? col : 0) * DIM;

    v8f acc = {};
    #pragma unroll 8
    for (int k = 0; k < DIM; k += 4) {
        const int kk = k + half * 2;
        if ((k & 63) == 0)
            __builtin_prefetch(arow + k + 256, 0, 1);   // global_prefetch_b8
        v2f a = *(const v2f*)(arow + kk);
        v2f b = *(const v2f*)(brow + kk);
        // D = A(16x4) * B(4x16) + C, full f32
        acc = __builtin_amdgcn_wmma_f32_16x16x4_f32(
            false, a, false, b, (short)0, acc, false, false);
    }

    // Bias (per output column).
    float biasv = 0.0f;
    if (col < 2)          biasv = bc[col];
    else if (col < NOUT)  biasv = bb[col - 2];
    #pragma unroll
    for (int j = 0; j < 8; ++j) acc[j] += biasv;

    // Spill accumulator through LDS to transpose lane<->row.
    // D layout: VGPR j, lanes 0-15 -> row j, lanes 16-31 -> row j+8, col = lane&15.
    float* T = tileLds[wave];
    #pragma unroll
    for (int j = 0; j < 8; ++j)
        T[(j + half * 8) * 16 + col] = acc[j];
    __syncthreads();

    // Epilogue: lanes 0-15 each own one row of the tile.
    if (half == 0) {
        const int  r = col;
        const long g = row0 + r;
        const float* Tr = tileLds[wave] + r * 16;
        const float l0 = Tr[0], l1 = Tr[1];
        const float d0 = Tr[6], d1 = Tr[7], d2 = Tr[8], d3 = Tr[9];

        // 2-class softmax, class-1 probability.
        const float score = 1.0f / (1.0f + expf(l0 - l1));
        const float sc = (score > SCORE_T) ? score : 0.0f;

        // Box decode (weights 10,10,5,5; clip dw/dh; clamp to image).
        const float4 p = *(const float4*)(proposals + g * 4);
        const float w  = p.z - p.x, h = p.w - p.y;
        const float cx = p.x + 0.5f * w, cy = p.y + 0.5f * h;
        const float dx = d0 / 10.0f, dy = d1 / 10.0f;
        const float dw = fminf(d2 / 5.0f, CLIPV);
        const float dh = fminf(d3 / 5.0f, CLIPV);
        const float pcx = dx * w + cx, pcy = dy * h + cy;
        const float pw = expf(dw) * w, ph = expf(dh) * h;
        const float x1 = fminf(fmaxf(pcx - 0.5f * pw, 0.0f), IMG);
        const float y1 = fminf(fmaxf(pcy - 0.5f * ph, 0.0f), IMG);
        const float x2 = fminf(fmaxf(pcx + 0.5f * pw, 0.0f), IMG);
        const float y2 = fminf(fmaxf(pcy + 0.5f * ph, 0.0f), IMG);

        float* o = out + g * 5;
        o[0] = x1; o[1] = y1; o[2] = x2; o[3] = y2;
        ws_scores[g] = sc;
    }
}

// ---------------------------------------------------------------------------
// Kernel 2: per-batch NMS (sequential 100-iteration argmax/suppress, matching
// the reference scan semantics exactly, incl. tie-break = lowest index and
// unconditional keep[i]=valid / s[i]=0). One block per batch; scores, areas
// and keep flags live in LDS; boxes are re-read from d_out (L2-resident).
// ---------------------------------------------------------------------------
__global__ __launch_bounds__(256) void nms_kernel(
    const float* __restrict__ ws_scores,
    float*       __restrict__ out)
{
    __shared__ float         s[NPROP];
    __shared__ float         area[NPROP];
    __shared__ unsigned char keep[NPROP];
    __shared__ float         red_v[256];
    __shared__ int           red_i[256];

    const int  b    = blockIdx.x;
    const int  tid  = threadIdx.x;
    const long base = (long)b * NPROP;

    for (int n = tid; n < NPROP; n += 256) {
        s[n]    = ws_scores[base + n];
        keep[n] = 0;
        const float* o = out + (base + n) * 5;
        const float w = fmaxf(o[2] - o[0], 0.0f);
        const float h = fmaxf(o[3] - o[1], 0.0f);
        area[n] = w * h;
    }
    __syncthreads();

    for (int it = 0; it < MAXDET; ++it) {
        // --- argmax over s[0..NPROP), first-max-index semantics ---
        float bv = -1.0f; int bi = 0;
        for (int n = tid; n < NPROP; n += 256) {
            const float v = s[n];
            if (v > bv) { bv = v; bi = n; }        // strided scan is ascending
        }
        red_v[tid] = bv; red_i[tid] = bi;
        __syncthreads();
        for (int off = 128; off > 0; off >>= 1) {
            if (tid < off) {
                const float ov = red_v[tid + off];
                const int   oi = red_i[tid + off];
                if (ov > red_v[tid] || (ov == red_v[tid] && oi < red_i[tid])) {
                    red_v[tid] = ov; red_i[tid] = oi;
                }
            }
            __syncthreads();
        }
        const int   i     = red_i[0];
        const float vmax  = red_v[0];
        const bool  valid = vmax > 0.0f;

        const float* bo = out + (base + i) * 5;
        const float bx1 = bo[0], by1 = bo[1], bx2 = bo[2], by2 = bo[3];
        const float ai  = area[i];

        // --- suppress ---
        for (int n = tid; n < NPROP; n += 256) {
            const float* on = out + (base + n) * 5;
            const float ix1 = fmaxf(bx1, on[0]);
            const float iy1 = fmaxf(by1, on[1]);
            const float ix2 = fminf(bx2, on[2]);
            const float iy2 = fminf(by2, on[3]);
            const float inter = fmaxf(ix2 - ix1, 0.0f) * fmaxf(iy2 - iy1, 0.0f);
            const float iou   = inter / (ai + area[n] - inter + 1e-9f);
            if (n == i) {
                s[n]    = 0.0f;                    // .at[i].set(0.0)
                keep[n] = valid ? 1 : 0;           // .at[i].set(valid)
            } else if (iou > NMS_T) {
                s[n] = 0.0f;
            }
        }
        __syncthreads();   // also orders red_* reads vs next iteration's writes
    }

    for (int n = tid; n < NPROP; n += 256)
        out[(base + n) * 5 + 4] = ws_scores[base + n] * (keep[n] ? 1.0f : 0.0f);
}

// ---------------------------------------------------------------------------
extern "C" void kernel_launch(void* const* d_in, const int* in_sizes, int n_in,
                              void* d_out, int out_size, void* d_ws, size_t ws_size,
                              hipStream_t stream) {
    const float* feats     = (const float*)d_in[0];   // (16,4000,1024)
    const float* proposals = (const float*)d_in[1];   // (16,4000,4)
    const float* Wc        = (const float*)d_in[2];   // (2,1024)
    const float* bc        = (const float*)d_in[3];   // (2,)
    const float* Wb        = (const float*)d_in[4];   // (8,1024)
    const float* bb        = (const float*)d_in[5];   // (8,)
    float* out       = (float*)d_out;                 // (16,4000,5)
    float* ws_scores = (float*)d_ws;                  // 64000 floats scratch

    // 4000 row-tiles of 16 -> 1000 blocks x 4 waves.
    det_head_kernel<<<dim3(1000), dim3(128), 0, stream>>>(
        feats, proposals, Wc, bc, Wb, bb, out, ws_scores);
    nms_kernel<<<dim3(BATCH), dim3(256), 0, stream>>>(ws_scores, out);
}